// ContourDec_11759620456532
// MI455X (gfx1250) — compile-verified
//
#include <hip/hip_runtime.h>

typedef __attribute__((ext_vector_type(2))) float v2f;
typedef __attribute__((ext_vector_type(4))) float v4f;
typedef __attribute__((ext_vector_type(8))) float v8f;

// DFB filter f (12 taps), after sign flip of even indices (runtime-indexed -> constant mem)
__constant__ float F12[12] = {0.0144f, 0.0272f, 0.0526f, 0.0972f, 0.193f, 0.63f,
                              -0.63f, -0.193f, -0.0972f, -0.0526f, -0.0272f, -0.0144f};

#define SQRT2F    1.41421356237309515f
#define INVSQRT2F 0.70710678118654757f

// -------- Stage A: Laplacian lowpass (decimated separable h-filter) --------
// t1[n,i,j] = sum_a h[a] * x[n, (2i+a-4)%256, j]   (n<512, i<128, j<256)
// Vectorized float4 along j: B128 loads/stores for HBM-width streaming.
__global__ void k_lap_rowdec(const float* __restrict__ x, float* __restrict__ t1) {
  const float H[9] = {0.037828455506995f, -0.02384946501938f, -0.11062440441842f,
                      0.37740285561265f, 0.8526986790094f, 0.37740285561265f,
                      -0.11062440441842f, -0.02384946501938f, 0.037828455506995f};
  int idx = blockIdx.x * blockDim.x + threadIdx.x;       // 512*128*64
  int j4 = idx & 63;
  int i = (idx >> 6) & 127;
  int n = idx >> 13;
  const v4f* xp = (const v4f*)(x + ((size_t)n << 16));
  v4f s = {0.f, 0.f, 0.f, 0.f};
#pragma unroll
  for (int a = 0; a < 9; ++a) {
    int r = (2 * i + a - 4) & 255;
    v4f vv = xp[(r << 6) | j4];
    s.x += H[a] * vv.x;
    s.y += H[a] * vv.y;
    s.z += H[a] * vv.z;
    s.w += H[a] * vv.w;
  }
  ((v4f*)t1)[idx] = s;
}

// c[n,i,jj] = sum_b h[b] * t1[n, i, (2jj+b-4)%256]  -> out slice 0
__global__ void k_lap_coldec(const float* __restrict__ t1, float* __restrict__ cOut) {
  const float H[9] = {0.037828455506995f, -0.02384946501938f, -0.11062440441842f,
                      0.37740285561265f, 0.8526986790094f, 0.37740285561265f,
                      -0.11062440441842f, -0.02384946501938f, 0.037828455506995f};
  int idx = blockIdx.x * blockDim.x + threadIdx.x;       // 512*128*128
  int jj = idx & 127;
  int i = (idx >> 7) & 127;
  int n = idx >> 14;
  const float* tp = t1 + (((size_t)n * 128 + i) << 8);
  float s = 0.f;
#pragma unroll
  for (int b = 0; b < 9; ++b) {
    int cpos = (2 * jj + b - 4) & 255;
    s += H[b] * tp[cpos];
  }
  cOut[idx] = s;
}

// d(r,s) = sum_{a,b with parity} g[a]g[b] * c[((r+a-3)%256)/2, ((s+b-3)%256)/2]
// Fully unrolled 4x4 with parity-selected weights (zero-padded 3-tap case).
__device__ __forceinline__ float lap_d(const float* __restrict__ cp, int r, int s) {
  // even taps a = 0,2,4,6 ; odd taps a = 1,3,5,(pad 0)
  const float GE[4] = {-0.064538882628938f, 0.41809227322221f,
                       0.41809227322221f, -0.064538882628938f};
  const float GO[4] = {-0.040689417609558f, 0.78848561640566f,
                       -0.040689417609558f, 0.f};
  int rodd = r & 1;           // r even -> odd taps ; r odd -> even taps
  int sodd = s & 1;
  int pa = rodd ? 0 : 1;      // first tap index
  int pb = sodd ? 0 : 1;
  float wr[4], wc[4];
  int ri[4], si[4];
#pragma unroll
  for (int k = 0; k < 4; ++k) {
    wr[k] = rodd ? GE[k] : GO[k];
    wc[k] = sodd ? GE[k] : GO[k];
    ri[k] = ((r + pa + 2 * k - 3) & 255) >> 1;
    si[k] = ((s + pb + 2 * k - 3) & 255) >> 1;
  }
  float acc = 0.f;
#pragma unroll
  for (int k = 0; k < 4; ++k) {
    float rowsum = 0.f;
#pragma unroll
    for (int l = 0; l < 4; ++l)
      rowsum += wc[l] * cp[(ri[k] << 7) | si[l]];
    acc += wr[k] * rowsum;
  }
  return acc;
}

// -------- Stage B: fused highpass + quincunx '1r' gather --------
// p0[n,i,j] = xhi[n,(i-j)%256,(i+j)%256], p1 at col+1; xhi = x - d (fused)
__global__ void k_qp1(const float* __restrict__ x, const float* __restrict__ c,
                      float* __restrict__ p0, float* __restrict__ p1) {
  int idx = blockIdx.x * blockDim.x + threadIdx.x;       // 512*128*256
  int j = idx & 255;
  int i = (idx >> 8) & 127;
  int n = idx >> 15;
  int r  = (i - j) & 255;
  int s0 = (i + j) & 255;
  int s1 = (s0 + 1) & 255;
  const float* xp = x + ((size_t)n << 16);
  const float* cp = c + ((size_t)n << 14);
  p0[idx] = xp[(r << 8) | s0] - lap_d(cp, r, s0);
  p1[idx] = xp[(r << 8) | s1] - lap_d(cp, r, s1);
}

// -------- WMMA banded-matrix separable 12-tap filter passes --------
__device__ __forceinline__ float bandF(int d) {
  return (d >= 0 && d < 12) ? F12[d] : 0.0f;
}

// Vertical pass: dst[n,i,c] = sum_a f[a] * E(i+a-ru, c)
// E: periodic row wrap (M=128 -> & 127, branchless); qper XORs column with N/2
// on wrap via cndmask (no exec divergence).
// Tile: out(16x16) = BandA(16xK) * InWindow(Kx16), K = 28 (7 wmma k-steps).
__global__ void k_fvert_wmma(const float* __restrict__ src, float* __restrict__ dst,
                             int lgN, int ru, int qper) {
  const int N = 1 << lgN;
  int lane = threadIdx.x & 31;
  int mrow = lane & 15;
  int hi = lane >> 4;                      // 0 or 1
  // hoisted band A fragments (lane/chunk only; tile-invariant)
  float bax[7], bay[7];
#pragma unroll
  for (int t = 0; t < 7; ++t) {
    int ka = 4 * t + 2 * hi;
    bax[t] = bandF(ka - mrow);
    bay[t] = bandF(ka + 1 - mrow);
  }
  int n = blockIdx.y;
  int tileInImg = blockIdx.x * (blockDim.x >> 5) + (threadIdx.x >> 5);
  int lgTc = lgN - 4;
  int rowbase = (tileInImg >> lgTc) << 4;
  int colbase = (tileInImg & ((1 << lgTc) - 1)) << 4;
  const float* sp = src + ((size_t)n << (7 + lgN));
  int col = colbase + mrow;                // B/D column = lane&15
  int xorv = qper ? (N >> 1) : 0;          // uniform
  int base = rowbase - ru;                 // window start row (may be negative)
  v8f acc = {0.f, 0.f, 0.f, 0.f, 0.f, 0.f, 0.f, 0.f};
#pragma unroll
  for (int t = 0; t < 7; ++t) {
    v2f a;
    a.x = bax[t];
    a.y = bay[t];
    int kb = 4 * t + hi;
    v2f b;
    {
      int rr = base + kb;
      int cc = col ^ (((unsigned)rr >= 128u) ? xorv : 0);  // branchless qper roll
      b.x = sp[((rr & 127) << lgN) + cc];
    }
    {
      int rr = base + kb + 2;
      int cc = col ^ (((unsigned)rr >= 128u) ? xorv : 0);
      b.y = sp[((rr & 127) << lgN) + cc];
    }
    acc = __builtin_amdgcn_wmma_f32_16x16x4_f32(false, a, false, b,
                                                (short)0, acc, false, false);
  }
  float* dp = dst + ((size_t)n << (7 + lgN));
  int rbase = rowbase + 8 * hi;
#pragma unroll
  for (int v = 0; v < 8; ++v)
    dp[((rbase + v) << lgN) + col] = acc[v];
}

// Horizontal pass + combine: s = sum_b f[b] * tin[n,i,(j+b-cl)%N]
// val = cA*aux + cB*s ; optional ws store (dst) and/or scatter to out slices.
// Tile: out(16x16) = InWindow(16xK) * BandB(Kx16), K = 28.
__global__ void k_fhor_wmma(const float* __restrict__ tin, const float* __restrict__ aux,
                            float* __restrict__ dst, float* __restrict__ outp,
                            int lgN, int cl, float cA, float cB,
                            int sliceLo, int sliceHi) {
  const int N = 1 << lgN;
  int lane = threadIdx.x & 31;
  int mrow = lane & 15;
  int hi = lane >> 4;
  // hoisted band B fragments
  float bbx[7], bby[7];
#pragma unroll
  for (int t = 0; t < 7; ++t) {
    int kb = 4 * t + hi;
    bbx[t] = bandF(kb - mrow);
    bby[t] = bandF(kb + 2 - mrow);
  }
  int n = blockIdx.y;
  int tileInImg = blockIdx.x * (blockDim.x >> 5) + (threadIdx.x >> 5);
  int lgTc = lgN - 4;
  int rowbase = (tileInImg >> lgTc) << 4;
  int colbase = (tileInImg & ((1 << lgTc) - 1)) << 4;
  const float* tp = tin + ((size_t)n << (7 + lgN));
  int arow = (rowbase + mrow) << lgN;
  int cbase = colbase - cl;
  v8f acc = {0.f, 0.f, 0.f, 0.f, 0.f, 0.f, 0.f, 0.f};
#pragma unroll
  for (int t = 0; t < 7; ++t) {
    int ka = 4 * t + 2 * hi;
    v2f a;
    a.x = tp[arow + ((cbase + ka) & (N - 1))];
    a.y = tp[arow + ((cbase + ka + 1) & (N - 1))];
    v2f b;
    b.x = bbx[t];
    b.y = bby[t];
    acc = __builtin_amdgcn_wmma_f32_16x16x4_f32(false, a, false, b,
                                                (short)0, acc, false, false);
  }
  int rbase = rowbase + 8 * hi;
  int jcol = colbase + mrow;
#pragma unroll
  for (int v = 0; v < 8; ++v) {
    int row = rbase + v;
    size_t idx = ((size_t)n << (7 + lgN)) + ((size_t)row << lgN) + jcol;
    float val = cA * aux[idx] + cB * acc[v];
    if (dst) dst[idx] = val;
    if (outp) {  // stage-2 only: N==128, chans-per-image = 128
      int img = n >> 7, c2 = n & 127;
      int slice = (c2 < 64) ? sliceLo : sliceHi;
      int cc = c2 & 63;
      size_t o = (size_t)slice * 8388608u +
                 ((((size_t)img * 64 + cc) * 128 + row) << 7) + jcol;
      outp[o] = val;
    }
  }
}

// -------- Stage C: quincunx '2c' gather over channel-concat(Y0,Y1) --------
__global__ void k_qp2(const float* __restrict__ y0, const float* __restrict__ y1,
                      float* __restrict__ p0c, float* __restrict__ p1c) {
  int idx = blockIdx.x * blockDim.x + threadIdx.x;       // 1024*128*128
  int j = idx & 127;
  int i = (idx >> 7) & 127;
  int n2 = idx >> 14;
  int img = n2 >> 7, c2 = n2 & 127;
  int colc = (j - i) & 255;
  int r0 = (i + j) & 127;
  int r1 = (i + j + 1) & 127;
  const float* base = (c2 < 64) ? y0 : y1;
  const float* sp = base + ((((size_t)img * 64 + (c2 & 63)) * 128) << 8);
  p0c[idx] = sp[(r0 << 8) | colc];
  p1c[idx] = sp[(r1 << 8) | colc];
}

extern "C" void kernel_launch(void* const* d_in, const int* in_sizes, int n_in,
                              void* d_out, int out_size, void* d_ws, size_t ws_size,
                              hipStream_t stream) {
  (void)in_sizes; (void)n_in; (void)out_size; (void)ws_size;
  const float* x = (const float*)d_in[0];
  float* out = (float*)d_out;
  float* ws  = (float*)d_ws;

  const size_t SLOT = 16777216;   // 64 MiB slots; total ws use = 5 slots = 320 MiB
  float* T  = ws;                 // filter temp (also t1)
  float* P0 = ws + SLOT;          // p0, later p0c
  float* P1 = ws + 2 * SLOT;      // p1, later p1c
  float* Y0 = ws + 3 * SLOT;      // y0, later y0c
  float* Y1 = ws + 4 * SLOT;      // y1

  dim3 blk(256);                  // 8 waves/block
  dim3 g1(16, 512);               // stage 1: 128 tiles/img (8x16), 512 images
  dim3 g2(8, 1024);               // stage 2: 64 tiles/img (8x8), 1024 images

  // Laplacian lowpass -> c (output slice 0)
  k_lap_rowdec<<<16384, blk, 0, stream>>>(x, T);      // float4: 512*128*64 threads
  k_lap_coldec<<<32768, blk, 0, stream>>>(T, out);

  // fused highpass + quincunx '1r'
  k_qp1<<<65536, blk, 0, stream>>>(x, out, P0, P1);

  // fbdec level 1 (qper_col): y0 = (p0 - F*p1)/sqrt2 ; y1 = -sqrt2*p1 - F*y0
  k_fvert_wmma<<<g1, blk, 0, stream>>>(P1, T, 8, 6, 1);
  k_fhor_wmma<<<g1, blk, 0, stream>>>(T, P0, Y0, nullptr, 8, 6,
                                      INVSQRT2F, -INVSQRT2F, 0, 0);
  k_fvert_wmma<<<g1, blk, 0, stream>>>(Y0, T, 8, 5, 1);
  k_fhor_wmma<<<g1, blk, 0, stream>>>(T, P1, Y1, nullptr, 8, 5,
                                      -SQRT2F, -1.0f, 0, 0);

  // quincunx '2c' over concat(Y0,Y1) -> P0c,P1c
  k_qp2<<<65536, blk, 0, stream>>>(Y0, Y1, P0, P1);

  // fbdec level 2 (per): y0c -> slices 2/3 (and ws for next pass), y1c -> slices 1/4
  k_fvert_wmma<<<g2, blk, 0, stream>>>(P1, T, 7, 6, 0);
  k_fhor_wmma<<<g2, blk, 0, stream>>>(T, P0, Y0, out, 7, 6,
                                      INVSQRT2F, -INVSQRT2F, 2, 3);
  k_fvert_wmma<<<g2, blk, 0, stream>>>(Y0, T, 7, 5, 0);
  k_fhor_wmma<<<g2, blk, 0, stream>>>(T, P1, nullptr, out, 7, 5,
                                      -SQRT2F, -1.0f, 1, 4);
}